// SemanticStructureFusionWeighter_47854525612391
// MI455X (gfx1250) — compile-verified
//
#include <hip/hip_runtime.h>
#include <hip/hip_bf16.h>

// Problem constants (match reference)
#define B_ 64
#define R_ 32
#define N_ 16384
#define D_ 64
#define O_ 8

#define SPLIT 16   // blocks per batch in stage 1
#define WAVES 8    // waves per block (256 threads, wave32)

typedef __attribute__((ext_vector_type(2))) float v2f;
typedef __attribute__((ext_vector_type(8))) float v8f;

// ---------------------------------------------------------------------------
// Stage 1: s[b,d] = sum_n fmask[b,n] * h_context[b,n,d] / ||h_context[b,n,:]||
//          cnt[b] = sum_n fmask[b,n]
// Mask density ~0.74%, so we ballot the mask for 32 rows at a time and only
// touch h_context for surviving rows (skips ~99.3% of the 256MB tensor).
// ---------------------------------------------------------------------------
__global__ void ssfw_stage1(const float* __restrict__ ctx,
                            const int* __restrict__ center_o,
                            const int* __restrict__ o_types,
                            const unsigned char* __restrict__ adj,
                            const unsigned char* __restrict__ two,
                            float* __restrict__ s_acc,   // [B][D]
                            float* __restrict__ cnt)     // [B]
{
    const int b    = blockIdx.x / SPLIT;
    const int seg  = blockIdx.x % SPLIT;
    const int wave = threadIdx.x >> 5;
    const int lane = threadIdx.x & 31;

    const int rowsPerBlock = N_ / SPLIT;            // 1024
    const int rowsPerWave  = rowsPerBlock / WAVES;  // 128
    const int base         = seg * rowsPerBlock + wave * rowsPerWave;

    const int co = center_o[b];
    const float2* __restrict__ ctx2 =
        (const float2*)ctx + (size_t)b * N_ * (D_ / 2);

    float2 acc = make_float2(0.f, 0.f);
    int count = 0;

    for (int c = 0; c < rowsPerWave; c += 32) {
        const int n = base + c + lane;
        const size_t idx = (size_t)b * N_ + n;
        const bool m = ((adj[idx] | two[idx]) != 0) && (o_types[idx] == co);

        unsigned mask = __builtin_amdgcn_ballot_w32(m);
        count += __popc(mask);

        while (mask) {
            const int j = __ffs(mask) - 1;
            mask &= mask - 1;
            const int nr = base + c + j;
            // lane l owns dims 2l, 2l+1 -> coalesced 256B load per wave
            const float2 x = ctx2[(size_t)nr * (D_ / 2) + lane];
            float sq = x.x * x.x + x.y * x.y;
            // wave32 tree reduction for the row's squared norm
            #pragma unroll
            for (int off = 16; off >= 1; off >>= 1)
                sq += __shfl_xor(sq, off, 32);
            const float inv = __frsqrt_rn(fmaxf(sq, 1e-12f));
            acc.x = fmaf(x.x, inv, acc.x);
            acc.y = fmaf(x.y, inv, acc.y);
        }
    }

    atomicAdd(&s_acc[b * D_ + 2 * lane + 0], acc.x);
    atomicAdd(&s_acc[b * D_ + 2 * lane + 1], acc.y);
    if (lane == 0) atomicAdd(&cnt[b], (float)count);
}

// ---------------------------------------------------------------------------
// Stage 2: per batch, raw[r] = l_local[b,r,:] . s[b,:] via V_WMMA_F32_16X16X4_F32
// (two 16-row tiles, 16 K-steps of 4; s replicated into all 16 B columns so
// every column of D equals the dot product). Row normalization of l is a
// per-row scalar -> folded in after the matmul.
//   w[b,r] = relu(lambda[r,co]/max(nv,1)) * (1 - raw[r]*invnorm_l[r]/max(nv,1e-9))
// One wave (32 threads) per batch; EXEC all-1s during WMMA as required.
// ---------------------------------------------------------------------------
__global__ void ssfw_stage2(const float* __restrict__ l_local,
                            const float* __restrict__ lambda_so,
                            const int* __restrict__ center_o,
                            const float* __restrict__ s_acc,
                            const float* __restrict__ cnt,
                            float* __restrict__ out)
{
    const int b    = blockIdx.x;
    const int lane = threadIdx.x;          // 0..31, one wave
    const int hi   = lane >> 4;            // lane half
    const int m16  = lane & 15;

    __shared__ float sh_s[D_];
    __shared__ float sh_inv[R_];

    // stage-1 result into LDS
    sh_s[2 * lane + 0] = s_acc[b * D_ + 2 * lane + 0];
    sh_s[2 * lane + 1] = s_acc[b * D_ + 2 * lane + 1];

    // invnorm of each l_local row (lane == row)
    const float* __restrict__ Lb = l_local + (size_t)b * R_ * D_;
    float sq = 0.f;
    #pragma unroll 8
    for (int d = 0; d < D_; ++d) {
        const float x = Lb[lane * D_ + d];
        sq = fmaf(x, x, sq);
    }
    sh_inv[lane] = __frsqrt_rn(fmaxf(sq, 1e-12f));
    __syncthreads();

    // WMMA accumulation: A = l rows (raw), B = s replicated across 16 cols.
    // A 16x4 layout: lanes 0-15 hold K=k0,k0+1; lanes 16-31 hold K=k0+2,k0+3.
    // B 4x16 layout mirrors the K split across lane halves.
    v8f c0 = {};   // rows 0..15
    v8f c1 = {};   // rows 16..31
    #pragma unroll
    for (int kc = 0; kc < 16; ++kc) {
        const int k0 = 4 * kc + 2 * hi;
        v2f bfrag; bfrag[0] = sh_s[k0];               bfrag[1] = sh_s[k0 + 1];
        v2f a0;    a0[0] = Lb[m16 * D_ + k0];         a0[1] = Lb[m16 * D_ + k0 + 1];
        v2f a1;    a1[0] = Lb[(16 + m16) * D_ + k0];  a1[1] = Lb[(16 + m16) * D_ + k0 + 1];
        c0 = __builtin_amdgcn_wmma_f32_16x16x4_f32(false, a0, false, bfrag,
                                                   (short)0, c0, false, false);
        c1 = __builtin_amdgcn_wmma_f32_16x16x4_f32(false, a1, false, bfrag,
                                                   (short)0, c1, false, false);
    }

    const float nv   = cnt[b];
    const float den1 = fmaxf(nv, 1e-9f);
    const float den2 = fmaxf(nv, 1.0f);
    const int   co   = center_o[b];

    // C/D layout: VGPR i = row i (lanes 0-15) / row i+8 (lanes 16-31); every
    // column is identical by construction, so lanes with m16==0 write out.
    if (m16 == 0) {
        const int rb0 = hi ? 8 : 0;
        #pragma unroll
        for (int i = 0; i < 8; ++i) {
            {
                const int r = rb0 + i;                 // tile 0: rows 0..15
                const float avg = c0[i] * sh_inv[r] / den1;
                const float lam = lambda_so[r * O_ + co];
                out[b * R_ + r] = fmaxf(lam / den2, 0.f) * (1.f - avg);
            }
            {
                const int r = 16 + rb0 + i;            // tile 1: rows 16..31
                const float avg = c1[i] * sh_inv[r] / den1;
                const float lam = lambda_so[r * O_ + co];
                out[b * R_ + r] = fmaxf(lam / den2, 0.f) * (1.f - avg);
            }
        }
    }
}

// ---------------------------------------------------------------------------
extern "C" void kernel_launch(void* const* d_in, const int* in_sizes, int n_in,
                              void* d_out, int out_size, void* d_ws, size_t ws_size,
                              hipStream_t stream)
{
    const float*         l_local   = (const float*)d_in[0];         // [B,R,D]
    const float*         h_context = (const float*)d_in[1];         // [B,N,D]
    const float*         lambda_so = (const float*)d_in[2];         // [R,O]
    const int*           center_o  = (const int*)d_in[3];           // [B]
    const int*           o_types   = (const int*)d_in[4];           // [B,N]
    const unsigned char* adj_mask  = (const unsigned char*)d_in[5]; // [B,N] bool
    const unsigned char* two_hop   = (const unsigned char*)d_in[6]; // [B,N] bool
    float* out = (float*)d_out;                                     // [B,R]

    float* s_acc = (float*)d_ws;          // B*D floats
    float* cntp  = s_acc + B_ * D_;       // B floats

    // zero accumulators every launch (graph-capture legal, deterministic)
    hipMemsetAsync(d_ws, 0, (size_t)(B_ * D_ + B_) * sizeof(float), stream);

    ssfw_stage1<<<B_ * SPLIT, 32 * WAVES, 0, stream>>>(
        h_context, center_o, o_types, adj_mask, two_hop, s_acc, cntp);

    ssfw_stage2<<<B_, 32, 0, stream>>>(
        l_local, lambda_so, center_o, s_acc, cntp, out);
}